// HolographicLinear_43052752175189
// MI455X (gfx1250) — compile-verified
//
#include <hip/hip_runtime.h>

typedef __attribute__((ext_vector_type(2))) float        v2f;
typedef __attribute__((ext_vector_type(8))) float        v8f;
typedef __attribute__((ext_vector_type(4))) unsigned int v4u;
typedef __attribute__((ext_vector_type(4))) int          v4i;
typedef __attribute__((ext_vector_type(8))) int          v8i;

#define DIM     4096u
#define ROWS    16384u          // BATCH*SEQ
#define KBLK    64u             // K panel depth
#define MW      128u            // rows per workgroup
#define NW      256u            // cols per workgroup
#define XSTRIDE 68u             // padded LDS row stride (floats): 64 + 4 -> conflict-free A reads
#define XBUFSZ  (MW * XSTRIDE)  // floats per X panel buffer

#if __has_builtin(__builtin_amdgcn_tensor_load_to_lds)
#define HAVE_TDM 1
#else
#define HAVE_TDM 0
#endif

#if HAVE_TDM
// Issue one TDM DMA: X[row0 : row0+128, kb : kb+64] (f32) -> LDS at lds_off,
// written with +4-dword padding after every 64-dword row (stride 68 floats).
__device__ __forceinline__ void tdm_load_panel(unsigned lds_off, unsigned long long ga) {
  v4u g0;
  g0.x = 1u;                                               // count=1, user descriptor
  g0.y = lds_off;                                          // lds_addr (bytes)
  g0.z = (unsigned)ga;                                     // global_addr[31:0]
  g0.w = (unsigned)((ga >> 32) & 0x01FFFFFFull)            // global_addr[56:32]
       | (2u << 30);                                       // type = 2 ("image")
  v8i g1;
  g1[0] = (int)((2u << 16)                                 // data_size = 4 bytes
              | (1u << 20)                                 // pad_enable
              | (5u << 22)                                 // pad_interval: every 64 dwords
              | (3u << 25));                               // pad_amount: 4 dwords
  g1[1] = (int)(DIM << 16);                                // tensor_dim0 = 4096 (low 16b @ [63:48])
  g1[2] = (int)(ROWS << 16);                               // tensor_dim1 = 16384 (low 16b @ [95:80])
  g1[3] = (int)(64u << 16);                                // tile_dim0 = 64 elements
  g1[4] = (int)MW;                                         // tile_dim1 = 128 rows, tile_dim2 = 0
  g1[5] = (int)DIM;                                        // tensor_dim0_stride = 4096
  g1[6] = 0;                                               // stride hi / dim1_stride lo
  g1[7] = 0;
  v4i z4 = {0, 0, 0, 0};                                   // groups 2/3 unused (2D tile)
  v8i z8 = {0, 0, 0, 0, 0, 0, 0, 0};                       // 6-arg form: extra (unused) group
  __builtin_amdgcn_tensor_load_to_lds(g0, g1, z4, z4, z8, 0);
}
#endif

// out[r, n] = sum_k x[r, k] * wr[(n - k) & 4095] + bias[n]
// (mag*cos(phase) == Re(ifft(fft(x)*fft(w))) => circulant GEMM; weight_imag drops out)
__global__ __launch_bounds__(256)
void holo_circconv_wmma(const float* __restrict__ x,
                        const float* __restrict__ wr,
                        const float* __restrict__ bias,
                        float* __restrict__ out) {
  __shared__ __align__(16) float wrS2[2u * DIM];   // doubled weight: no modulo in hot loop (32 KB)
  __shared__ __align__(16) float xS[2][XBUFSZ];    // double-buffered padded X panel (2 x 34 KB)

  const unsigned tid  = threadIdx.x;
  const unsigned lane = tid & 31u;
  const unsigned wave = tid >> 5;      // 0..7
  const unsigned hi   = lane >> 4;     // half-wave select (K phase of fragments)
  const unsigned l16  = lane & 15u;

  const unsigned row0  = blockIdx.y * MW;
  const unsigned col0  = blockIdx.x * NW;
  const unsigned waveM = wave >> 2;                  // 0..1 : 64-row slab
  const unsigned waveN = wave & 3u;                  // 0..3 : 64-col slab
  const unsigned nBase = col0 + waveN * 64u + l16;   // lane's column in sub-tile 0

  // Stage doubled weight_real into LDS (reads hit L2 after first WG).
  for (unsigned i = tid; i < 2u * DIM; i += 256u) wrS2[i] = wr[i & (DIM - 1u)];

  // 4x4 register tile of 16x16 accumulators = 128 VGPRs (wave32 allows up to 1024).
  v8f c[4][4];
  #pragma unroll
  for (int mt = 0; mt < 4; ++mt)
    #pragma unroll
    for (int nt = 0; nt < 4; ++nt)
      c[mt][nt] = (v8f){};

  const unsigned long long gbase =
      (unsigned long long)(uintptr_t)x + (unsigned long long)row0 * DIM * 4u;
#if HAVE_TDM
  unsigned ldsX[2];
  ldsX[0] = (unsigned)(uintptr_t)(&xS[0][0]);
  ldsX[1] = (unsigned)(uintptr_t)(&xS[1][0]);
  if (wave == 0) tdm_load_panel(ldsX[0], gbase);     // prefetch panel 0
#endif

  const unsigned nKB = DIM / KBLK;                   // 64 K-panels
  for (unsigned ib = 0; ib < nKB; ++ib) {
    const unsigned cur = ib & 1u;

#if HAVE_TDM
    if (wave == 0) {
      if (ib + 1u < nKB) {
        tdm_load_panel(ldsX[cur ^ 1u], gbase + (unsigned long long)(ib + 1u) * KBLK * 4u);
        __builtin_amdgcn_s_wait_tensorcnt(1);        // panel `ib` landed (TDM is in-order)
      } else {
        __builtin_amdgcn_s_wait_tensorcnt(0);
      }
    }
    __syncthreads();                                 // publish panel `ib` to all waves
#else
    // Fallback: cooperative synchronous load, same padded layout.
    {
      const unsigned c4 = (tid & 15u) * 4u;          // 16 threads span 64 cols
      const unsigned rb = tid >> 4;                  // 16 row groups
      #pragma unroll
      for (unsigned rr = 0; rr < MW; rr += 16u) {
        const float4 v = *(const float4*)(x + (size_t)(row0 + rb + rr) * DIM + ib * KBLK + c4);
        *(float4*)(&xS[cur][(rb + rr) * XSTRIDE + c4]) = v;
      }
    }
    __syncthreads();
#endif

    const float* xbuf = &xS[cur][0];
    const unsigned kb = ib * KBLK;

    #pragma unroll 2
    for (unsigned k4 = 0; k4 < KBLK; k4 += 4u) {
      // A fragments (16x4 f32): lanes 0-15 K=k4,k4+1 ; lanes 16-31 K=k4+2,k4+3.
      // Stride 68 => banks (4*l16 + k) cover all 64 banks exactly once.
      v2f a[4];
      #pragma unroll
      for (int mt = 0; mt < 4; ++mt) {
        const unsigned r = waveM * 64u + (unsigned)mt * 16u + l16;
        a[mt] = *(const v2f*)(xbuf + r * XSTRIDE + k4 + 2u * hi);
      }

      // B fragments (4x16 f32): B[k][n] = wr[(n-k) mod D]; doubled table => no mask.
      const unsigned kA    = kb + k4 + 2u * hi;
      const unsigned jbase = nBase + DIM - kA;       // in [1, 8191]
      v2f b[4];
      #pragma unroll
      for (int nt = 0; nt < 4; ++nt) {
        const unsigned j = jbase + (unsigned)nt * 16u;
        b[nt].x = wrS2[j];
        b[nt].y = wrS2[j - 1u];
      }

      // 16 WMMAs per 12 LDS ops: 4x register reuse of both fragments.
      #pragma unroll
      for (int mt = 0; mt < 4; ++mt)
        #pragma unroll
        for (int nt = 0; nt < 4; ++nt)
          c[mt][nt] = __builtin_amdgcn_wmma_f32_16x16x4_f32(
              false, a[mt], false, b[nt], (short)0, c[mt][nt], false, false);
    }
    __syncthreads();                                 // everyone done with panel `ib`
  }

  // Epilogue: C VGPR r holds (M = r + 8*hi, N = l16) per 16x16 tile.
  #pragma unroll
  for (int nt = 0; nt < 4; ++nt) {
    const unsigned n  = nBase + (unsigned)nt * 16u;
    const float    bn = bias[n];
    #pragma unroll
    for (int mt = 0; mt < 4; ++mt) {
      const unsigned m0 = row0 + waveM * 64u + (unsigned)mt * 16u + hi * 8u;
      #pragma unroll
      for (int r = 0; r < 8; ++r)
        out[(size_t)(m0 + r) * DIM + n] = c[mt][nt][r] + bn;
    }
  }
}

extern "C" void kernel_launch(void* const* d_in, const int* in_sizes, int n_in,
                              void* d_out, int out_size, void* d_ws, size_t ws_size,
                              hipStream_t stream) {
  (void)in_sizes; (void)n_in; (void)d_ws; (void)ws_size; (void)out_size;
  const float* x    = (const float*)d_in[0];   // [4, 4096, 4096] f32
  const float* wr   = (const float*)d_in[1];   // [4096] f32
  // d_in[2] = weight_imag: output is Re(.), contributes nothing
  const float* bias = (const float*)d_in[3];   // [4096] f32
  float* out        = (float*)d_out;

  dim3 grid(DIM / NW, ROWS / MW);              // (16, 128) workgroups of 256 threads
  holo_circconv_wmma<<<grid, 256, 0, stream>>>(x, wr, bias, out);
}